// GateNetwork_1623497638568
// MI455X (gfx1250) — compile-verified
//
#include <hip/hip_runtime.h>
#include <math.h>

// Problem constants (from reference): x[B,D,L] fp32, W[E,D], b[E]
#define BATCH 1024
#define DIM   1024
#define SEQ   512
#define NEXP  16
#define NEG_SLOPE 0.2f

typedef __attribute__((ext_vector_type(2))) float v2f;
typedef __attribute__((ext_vector_type(4))) float v4f;
typedef __attribute__((ext_vector_type(8))) float v8f;

// ---------------------------------------------------------------------------
// Kernel 1: streaming pool (mean + max over L=512), bandwidth-bound (2 GiB).
// One wave32 per (b,d) row. Non-temporal b128 loads (read-once stream).
// Output is TRANSPOSED: pooledT[d*BATCH + b], so the WMMA B-fragment loads in
// kernel 2 are contiguous across lanes.
// ---------------------------------------------------------------------------
__global__ __launch_bounds__(256) void pool_kernel(const float* __restrict__ x,
                                                   float* __restrict__ pooledT) {
  const int lane = threadIdx.x & 31;
  // q indexes the *transposed* output: q = d*BATCH + b  (so the 8 waves of a
  // block write 8 consecutive floats of pooledT)
  const unsigned q = blockIdx.x * 8u + (threadIdx.x >> 5);
  const unsigned b = q & (BATCH - 1);
  const unsigned d = q >> 10;  // DIM == 1024
  const float* __restrict__ p = x + ((long long)b * DIM + d) * (long long)SEQ;

  float s = 0.0f;
  float m = -__builtin_inff();
#pragma unroll
  for (int i = 0; i < 4; ++i) {
    // 32 lanes * 16B = 512B contiguous per iteration; 4 iterations = 2KB row
    v4f v = __builtin_nontemporal_load((const v4f*)(p + i * 128 + lane * 4));
    s += v.x + v.y + v.z + v.w;
    m = fmaxf(m, fmaxf(fmaxf(v.x, v.y), fmaxf(v.z, v.w)));
  }
  // wave32 butterfly reduction
#pragma unroll
  for (int off = 16; off > 0; off >>= 1) {
    s += __shfl_xor(s, off, 32);
    m = fmaxf(m, __shfl_xor(m, off, 32));
  }
  if (lane == 0) pooledT[q] = s * (1.0f / (float)SEQ) + m;
}

// ---------------------------------------------------------------------------
// Kernel 2: logits via V_WMMA_F32_16X16X4_F32, then fused bias + LeakyReLU +
// top-2 + masked softmax. One wave per 16-row batch tile (64 waves total).
//
//   D[e][b] = sum_k W[e][k] * pooledT[k][b]
//
// A-fragment (16x4 f32, W tile): lane l<16 -> row e=l, K = k0+{0,1};
//                                lane l>=16 -> row e=l-16, K = k0+{2,3}
// B-fragment (4x16 f32, pooledT tile): VGPR0 lanes 0-15: K=k0,   N=b0+lane
//                                      VGPR0 lanes16-31: K=k0+2, N=b0+lane-16
//                                      VGPR1: K=k0+1 / K=k0+3
// D-matrix (16x16 f32): VGPR r, lanes 0-15: M=r, N=lane; lanes16-31: M=r+8.
// ---------------------------------------------------------------------------
__global__ __launch_bounds__(32) void gate_kernel(const float* __restrict__ pooledT,
                                                  const float* __restrict__ W,
                                                  const float* __restrict__ bias,
                                                  float* __restrict__ out) {
  const int lane = threadIdx.x & 31;
  const int half = lane >> 4;   // 0: lanes 0-15, 1: lanes 16-31
  const int l16  = lane & 15;
  const int b0   = blockIdx.x * 16;

  v8f acc = {};
  const float* wrow = W + l16 * DIM + 2 * half;                 // W[e=l16][k0+2*half ..]
  const float* pcol = pooledT + (2 * half) * BATCH + b0 + l16;  // pooledT[k0+2*half][b]

#pragma unroll 4
  for (int k0 = 0; k0 < DIM; k0 += 4) {
    v2f a;
    a.x = wrow[k0];
    a.y = wrow[k0 + 1];
    v2f bm;
    bm.x = pcol[k0 * BATCH];
    bm.y = pcol[(k0 + 1) * BATCH];
    // 8 args: (neg_a, A, neg_b, B, c_mod, C, reuse_a, reuse_b)
    acc = __builtin_amdgcn_wmma_f32_16x16x4_f32(false, a, false, bm,
                                                (short)0, acc, false, false);
  }

  // Gather all 16 expert logits for batch row (b0 + l16) into each lane.
  // Lane l and lane l^16 hold complementary expert halves of the same row.
  float vals[NEXP];
#pragma unroll
  for (int r = 0; r < 8; ++r) {
    float mine  = acc[r];
    float other = __shfl_xor(mine, 16, 32);
    vals[r]     = half ? other : mine;   // experts 0..7
    vals[r + 8] = half ? mine  : other;  // experts 8..15
  }

  // bias + LeakyReLU + top-2 (strict '>' == jax top_k lower-index tie-break)
  float top1 = -__builtin_inff(), top2 = -__builtin_inff();
  int i1 = 0, i2 = 0;
#pragma unroll
  for (int e = 0; e < NEXP; ++e) {
    float l = vals[e] + bias[e];
    l = (l >= 0.0f) ? l : NEG_SLOPE * l;
    if (l > top1) {
      top2 = top1; i2 = i1;
      top1 = l;    i1 = e;
    } else if (l > top2) {
      top2 = l; i2 = e;
    }
  }

  // softmax over the two kept logits (all others are exp(-inf) = 0)
  float e2    = __expf(top2 - top1);
  float denom = 1.0f + e2;
  float g1    = 1.0f / denom;
  float g2    = e2 / denom;

  // lanes 0-15 write gating rows; lanes 16-31 write mask rows
  const int brow = b0 + l16;
  float* dst = out + (half ? (BATCH * NEXP + brow * NEXP) : (brow * NEXP));
#pragma unroll
  for (int e = 0; e < NEXP; ++e) {
    float g  = (e == i1) ? g1 : ((e == i2) ? g2 : 0.0f);
    float mk = (e == i1 || e == i2) ? 1.0f : 0.0f;
    dst[e] = half ? mk : g;
  }
}

// ---------------------------------------------------------------------------
// Launch: inputs in reference setup_inputs() order: x, W, b. Output is
// (gating_coeffs [1024,16], mask [1024,16]) concatenated flat = 32768 f32.
// d_ws holds the 4 MB transposed pooled matrix.
// ---------------------------------------------------------------------------
extern "C" void kernel_launch(void* const* d_in, const int* in_sizes, int n_in,
                              void* d_out, int out_size, void* d_ws, size_t ws_size,
                              hipStream_t stream) {
  const float* x    = (const float*)d_in[0];
  const float* W    = (const float*)d_in[1];
  const float* bias = (const float*)d_in[2];
  float* out        = (float*)d_out;
  float* pooledT    = (float*)d_ws;  // [DIM][BATCH] = 4 MB

  const int n_rows = BATCH * DIM;            // 1,048,576 pooled values
  const int blocks = n_rows / 8;             // 8 waves (256 threads) per block
  pool_kernel<<<blocks, 256, 0, stream>>>(x, pooledT);

  gate_kernel<<<BATCH / 16, 32, 0, stream>>>(pooledT, W, bias, out);
}